// FairADG_6296422056683
// MI455X (gfx1250) — compile-verified
//
#include <hip/hip_runtime.h>
#include <hip/hip_bf16.h>
#include <stdint.h>

#define N_NODES 100000
#define E_EDGES 1600000
#define NFEAT   256
#define HIDDEN  256
#define C_GRP   4
#define PC      64
#define NCLASS  4

typedef __attribute__((ext_vector_type(2))) float v2f;
typedef __attribute__((ext_vector_type(8))) float v8f;

// ---------------------------------------------------------------------------
// Kernel A: proj[n][0..3] = x[n] . W1a[cc], proj[n][4..7] = x[n] . W1b[cc]
// wave-per-node, full-wave shuffle reduction.
// ---------------------------------------------------------------------------
__global__ void proj_kernel(const float* __restrict__ x,
                            const float* __restrict__ asg_W1, // (4, 512)
                            float* __restrict__ proj)         // (N, 8)
{
    int node = (int)((blockIdx.x * blockDim.x + threadIdx.x) >> 5);
    int lane = threadIdx.x & 31;
    if (node >= N_NODES) return;

    const float* xr = x + (size_t)node * NFEAT + lane * 8;
    float xv[8];
#pragma unroll
    for (int j = 0; j < 8; ++j) xv[j] = xr[j];

    float partial[8];
#pragma unroll
    for (int o = 0; o < 8; ++o) {
        int cc = o & 3, half = o >> 2;
        const float* wr = asg_W1 + cc * (2 * NFEAT) + half * NFEAT + lane * 8;
        float s = 0.f;
#pragma unroll
        for (int j = 0; j < 8; ++j) s += xv[j] * wr[j];
        // full-wave (32 lane) tree reduction
        s += __shfl_xor(s, 1, 32);
        s += __shfl_xor(s, 2, 32);
        s += __shfl_xor(s, 4, 32);
        s += __shfl_xor(s, 8, 32);
        s += __shfl_xor(s, 16, 32);
        partial[o] = s;
    }
    if (lane == 0) {
#pragma unroll
        for (int o = 0; o < 8; ++o) proj[(size_t)node * 8 + o] = partial[o];
    }
}

// ---------------------------------------------------------------------------
// Kernel B: z = x @ linW^T + lin_b ; c = per-group z @ convW^T  (f32 WMMA)
// block = 128 threads = 4 waves; each wave owns a 16-node tile (64/block).
// Dynamic LDS: x tile [64][260] + z tile [64][260]  (stride 260 vs bank=64)
// x tile is filled with CDNA5 async cache->LDS copies (ASYNCcnt), not VGPRs.
// ---------------------------------------------------------------------------
__global__ void zc_kernel(const float* __restrict__ x,
                          const float* __restrict__ linW,  // (256,256) = lin_W flat
                          const float* __restrict__ linB,  // 256
                          const float* __restrict__ convW, // (256,64)  = conv_W flat
                          float* __restrict__ c_out)       // (N,256)
{
    extern __shared__ float lds[];
    const int STR = 260;
    float* lds_x = lds;
    float* lds_z = lds + 64 * STR;

    int tid  = threadIdx.x;     // 0..127
    int wave = tid >> 5;        // 0..3
    int lane = tid & 31;
    int node_base = blockIdx.x * 64;

    // phase 1: async cache->LDS fill of the 64x256 x tile (zero-fill past N)
    for (int i = tid; i < 64 * (NFEAT / 4); i += 128) {
        int r  = i / (NFEAT / 4);
        int cq = (i % (NFEAT / 4)) * 4;
        int node = node_base + r;
        float* ldst = lds_x + r * STR + cq;
        if (node < N_NODES) {
            const float* gsrc = x + (size_t)node * NFEAT + cq;
            unsigned lds_off = (unsigned)(uintptr_t)ldst;  // low 32b = LDS byte addr
            asm volatile("global_load_async_to_lds_b128 %0, %1, off"
                         :
                         : "v"(lds_off), "v"(gsrc)
                         : "memory");
        } else {
            ldst[0] = 0.f; ldst[1] = 0.f; ldst[2] = 0.f; ldst[3] = 0.f;
        }
    }
    asm volatile("s_wait_asynccnt 0x0" ::: "memory");
    __syncthreads();

    int half = lane >> 4;              // A/B K-half per ISA 16x4 f32 layout
    int l15  = lane & 15;
    int rowA = wave * 16 + l15;

    // phase 2: z tile via V_WMMA_F32_16X16X4_F32, K-blocked over 256 feats
    for (int nt = 0; nt < 16; ++nt) {
        float bb = linB[nt * 16 + l15];
        v8f acc = {bb, bb, bb, bb, bb, bb, bb, bb};
        const float* brow = linW + (size_t)(nt * 16 + l15) * NFEAT;
        for (int k0 = 0; k0 < NFEAT; k0 += 4) {
            v2f a = *(const v2f*)(lds_x + rowA * STR + k0 + 2 * half);
            v2f b = *(const v2f*)(brow + k0 + 2 * half);
            acc = __builtin_amdgcn_wmma_f32_16x16x4_f32(
                false, a, false, b, (short)0, acc, false, false);
        }
        // D layout: VGPR v -> row v (lanes 0-15) / v+8 (lanes 16-31), col = l15
#pragma unroll
        for (int v = 0; v < 8; ++v) {
            int r = wave * 16 + v + 8 * half;
            lds_z[r * STR + nt * 16 + l15] = acc[v];
        }
    }
    __syncthreads();

    // phase 3: c[n, g*64+q] = sum_p z[n, g*64+p] * convW[(g*64+q)*64+p]
    for (int g = 0; g < C_GRP; ++g) {
        for (int nt = 0; nt < 4; ++nt) {
            v8f acc = {0.f, 0.f, 0.f, 0.f, 0.f, 0.f, 0.f, 0.f};
            const float* brow = convW + (size_t)(g * 64 + nt * 16 + l15) * PC;
            for (int p0 = 0; p0 < PC; p0 += 4) {
                v2f a = *(const v2f*)(lds_z + rowA * STR + g * 64 + p0 + 2 * half);
                v2f b = *(const v2f*)(brow + p0 + 2 * half);
                acc = __builtin_amdgcn_wmma_f32_16x16x4_f32(
                    false, a, false, b, (short)0, acc, false, false);
            }
#pragma unroll
            for (int v = 0; v < 8; ++v) {
                int node = node_base + wave * 16 + v + 8 * half;
                if (node < N_NODES)
                    c_out[(size_t)node * HIDDEN + g * 64 + nt * 16 + l15] = acc[v];
            }
        }
    }
}

// ---------------------------------------------------------------------------
// Kernel C: per-edge softmax gate + gather c[col] + atomic scatter to out[row]
// wave-per-edge; 8 floats per lane. c (100MB) is L2-resident on MI455X (192MB).
// Each lane prefetches its c[col] slice so the L2 gather overlaps gate math.
// ---------------------------------------------------------------------------
__global__ void edge_kernel(const long long* __restrict__ row,
                            const long long* __restrict__ col,
                            const float* __restrict__ proj,   // (N,8)
                            const float* __restrict__ asg_b1, // 4
                            const float* __restrict__ asg_W2, // (4,4)
                            const float* __restrict__ asg_b2, // 4
                            const float* __restrict__ c_in,   // (N,256)
                            float* __restrict__ out)          // (N,256)
{
    int e = blockIdx.x * (blockDim.x >> 5) + (threadIdx.x >> 5);
    int lane = threadIdx.x & 31;
    if (e >= E_EDGES) return;

    long long r  = row[e];
    long long cl = col[e];

    int   cbase = lane * 8;                                 // lane's 8 columns
    const float* cr = c_in + (size_t)cl * HIDDEN + cbase;
    __builtin_prefetch(cr, 0, 3);                           // global_prefetch_b8

    float h1[4];
#pragma unroll
    for (int i = 0; i < 4; ++i)
        h1[i] = proj[cl * 8 + i] + proj[r * 8 + 4 + i] + asg_b1[i];

    float h2[4], m = -1e30f;
#pragma unroll
    for (int j = 0; j < 4; ++j) {
        float s = asg_b2[j];
#pragma unroll
        for (int i = 0; i < 4; ++i) s += h1[i] * asg_W2[j * 4 + i];
        h2[j] = s;
        m = fmaxf(m, s);
    }
    float denom = 0.f;
#pragma unroll
    for (int j = 0; j < 4; ++j) { h2[j] = __expf(h2[j] - m); denom += h2[j]; }
    float inv = 1.f / denom;

    float w = h2[cbase >> 6] * inv;                         // group index = lane/8
    float* orow = out + (size_t)r * HIDDEN + cbase;
#pragma unroll
    for (int j = 0; j < 8; ++j) atomicAdd(orow + j, cr[j] * w);
}

// ---------------------------------------------------------------------------
// Kernel D: out += bias; per-group L2 normalize (in place); logits.
// wave-per-node; 8-lane octet reduce for group norms, 32-lane for logits.
// ---------------------------------------------------------------------------
__global__ void finalize_kernel(float* __restrict__ h,          // (N,256)
                                const float* __restrict__ bias, // 256
                                const float* __restrict__ clsW, // (4,256)
                                const float* __restrict__ clsB, // 4
                                float* __restrict__ logits)     // (N,4)
{
    int n = blockIdx.x * (blockDim.x >> 5) + (threadIdx.x >> 5);
    int lane = threadIdx.x & 31;
    if (n >= N_NODES) return;

    float* hr = h + (size_t)n * HIDDEN + lane * 8;
    float v[8];
    float ss = 0.f;
#pragma unroll
    for (int j = 0; j < 8; ++j) {
        v[j] = hr[j] + bias[lane * 8 + j];
        ss += v[j] * v[j];
    }
    // lanes 8g..8g+7 hold one 64-wide group: reduce within the octet
    ss += __shfl_xor(ss, 1, 32);
    ss += __shfl_xor(ss, 2, 32);
    ss += __shfl_xor(ss, 4, 32);
    float scale = 1.f / fmaxf(sqrtf(ss), 1e-12f);
#pragma unroll
    for (int j = 0; j < 8; ++j) { v[j] *= scale; hr[j] = v[j]; }

#pragma unroll
    for (int cls = 0; cls < NCLASS; ++cls) {
        float s = 0.f;
#pragma unroll
        for (int j = 0; j < 8; ++j)
            s += v[j] * clsW[cls * HIDDEN + lane * 8 + j];
        s += __shfl_xor(s, 1, 32);
        s += __shfl_xor(s, 2, 32);
        s += __shfl_xor(s, 4, 32);
        s += __shfl_xor(s, 8, 32);
        s += __shfl_xor(s, 16, 32);
        if (lane == 0) logits[(size_t)n * NCLASS + cls] = s + clsB[cls];
    }
}

// ---------------------------------------------------------------------------
extern "C" void kernel_launch(void* const* d_in, const int* in_sizes, int n_in,
                              void* d_out, int out_size, void* d_ws, size_t ws_size,
                              hipStream_t stream) {
    const float*     x       = (const float*)d_in[0];
    const long long* row     = (const long long*)d_in[1];  // int64 in reference
    const long long* col     = (const long long*)d_in[2];
    const float*     asg_W1  = (const float*)d_in[3];
    const float*     asg_b1  = (const float*)d_in[4];
    const float*     asg_W2  = (const float*)d_in[5];
    const float*     asg_b2  = (const float*)d_in[6];
    const float*     lin_W   = (const float*)d_in[7];
    const float*     lin_b   = (const float*)d_in[8];
    const float*     conv_W  = (const float*)d_in[9];
    const float*     bias    = (const float*)d_in[10];
    const float*     cls_W   = (const float*)d_in[11];
    const float*     cls_b   = (const float*)d_in[12];

    float* h      = (float*)d_out;                      // (N,256)
    float* logits = h + (size_t)N_NODES * HIDDEN;       // (N,4)

    float* proj  = (float*)d_ws;                        // (N,8)
    float* c_buf = proj + (size_t)N_NODES * 8;          // (N,256), ~100MB (L2-resident)

    // 1) node projections for the edge gate
    proj_kernel<<<(N_NODES + 7) / 8, 256, 0, stream>>>(x, asg_W1, proj);

    // 2) z and blocked conv via f32 WMMA -> c workspace
    size_t lds_bytes = (size_t)2 * 64 * 260 * sizeof(float);  // 133 KB of 320 KB WGP LDS
    zc_kernel<<<(N_NODES + 63) / 64, 128, lds_bytes, stream>>>(
        x, lin_W, lin_b, conv_W, c_buf);

    // 3) zero the segment-sum accumulator (h region of d_out)
    hipMemsetAsync(h, 0, (size_t)N_NODES * HIDDEN * sizeof(float), stream);

    // 4) edge gather/gate/atomic-scatter
    edge_kernel<<<(E_EDGES + 7) / 8, 256, 0, stream>>>(
        row, col, proj, asg_b1, asg_W2, asg_b2, c_buf, h);

    // 5) bias + group L2 norm + classifier
    finalize_kernel<<<(N_NODES + 7) / 8, 256, 0, stream>>>(
        h, bias, cls_W, cls_b, logits);
}